// GCN_52132313038906
// MI455X (gfx1250) — compile-verified
//
#include <hip/hip_runtime.h>

typedef float v2f __attribute__((ext_vector_type(2)));
typedef float v8f __attribute__((ext_vector_type(8)));

// -------------------------------------------------------------------------
// Dense GEMM C[M,N] = A[M,K] @ B[K,N], row-major FP32, V_WMMA_F32_16X16X4_F32.
// N = NT*16 is compile-time. One wave32 computes a full 16-row x N block.
// B is staged through LDS per K-chunk (KC) in a pair-swizzled layout so each
// B operand is one ds_load_b64; the A chunk is preloaded into registers as a
// single clause of global_load_b64 so memory latency is paid once per chunk.
//
// A-matrix VGPR layout (ISA 7.12.2, 32-bit A 16x4):
//   lanes 0-15 : M = lane,     VGPR0 = K+0, VGPR1 = K+1
//   lanes 16-31: M = lane-16,  VGPR0 = K+2, VGPR1 = K+3
// B-matrix (4x16) symmetric: N = lane&15, K offset 2*(lane>>4) + {0,1}.
// C/D: VGPR v -> M = v + 8*(lane>>4), N = lane&15.
//
// LDS layout (floats): lds[k4][h][n][j] = B[k0 + 4*k4 + 2*h + j][n]
//   read as v2f:  ((k4*2 + h)*N + n)  ->  {B[kk+2h][n], B[kk+2h+1][n]}
// -------------------------------------------------------------------------
template <int NT, int KC>
__global__ void __launch_bounds__(256)
gcn_gemm_wmma_lds(const float* __restrict__ A, const float* __restrict__ B,
                  float* __restrict__ C, int M, int K)
{
    constexpr int N = NT * 16;
    __shared__ float lds[(KC / 4) * 2 * N * 2];

    const int wid    = threadIdx.x >> 5;
    const int tm     = blockIdx.x * (blockDim.x >> 5) + wid;
    const bool active = (tm * 16 < M);        // wave-uniform

    const int lane = threadIdx.x & 31;
    const int half = lane >> 4;               // 0: K pair {0,1}, 1: {2,3}
    const int l16  = lane & 15;

    const float* arow = A + (size_t)(tm * 16 + l16) * K + 2 * half;

    v8f acc[NT];
    const v8f vzero = {};
#pragma unroll
    for (int t = 0; t < NT; ++t) acc[t] = vzero;

    for (int k0 = 0; k0 < K; k0 += KC) {
        __syncthreads();
        // ---- cooperative fill of B chunk into swizzled LDS -------------
        for (int idx = threadIdx.x; idx < KC * N; idx += 256) {
            const int kk = idx / N;           // N is a power of two -> shifts
            const int n  = idx - kk * N;
            const int k4 = kk >> 2;
            const int h  = (kk >> 1) & 1;
            const int j  = kk & 1;
            lds[(((k4 * 2 + h) * N) + n) * 2 + j] = B[(size_t)(k0 + kk) * N + n];
        }
        __syncthreads();

        if (active) {
            // ---- preload A chunk: one clause of b64 loads --------------
            v2f av[KC / 4];
#pragma unroll
            for (int i = 0; i < KC / 4; ++i) {
                av[i].x = arow[k0 + 4 * i];
                av[i].y = arow[k0 + 4 * i + 1];
            }
            // ---- compute: NT ds_load_b64 + NT WMMAs per K-step ---------
#pragma unroll
            for (int i = 0; i < KC / 4; ++i) {
                const v2f* bp = ((const v2f*)lds) + (i * 2 + half) * N + l16;
                v2f bv[NT];
#pragma unroll
                for (int t = 0; t < NT; ++t) bv[t] = bp[t * 16];
#pragma unroll
                for (int t = 0; t < NT; ++t)
                    acc[t] = __builtin_amdgcn_wmma_f32_16x16x4_f32(
                        /*neg_a=*/false, av[i], /*neg_b=*/false, bv[t],
                        /*c_mod=*/(short)0, acc[t],
                        /*reuse_a=*/false, /*reuse_b=*/false);
            }
        }
    }

    if (active) {
        float* crow = C + (size_t)(tm * 16 + half * 8) * N + l16;
#pragma unroll
        for (int t = 0; t < NT; ++t)
#pragma unroll
            for (int v = 0; v < 8; ++v)
                crow[(size_t)v * N + t * 16] = acc[t][v];
    }
}

// -------------------------------------------------------------------------
// SpMM scatter: dst[row[e], :] += w[e] * src[col[e], :]
// One thread per (edge, 4-feature group): float4 gather + 4 hardware FP32
// global atomics. Q = D/4 groups per edge; for D=128 one wave exactly covers
// one edge row (e = t>>5 wave-uniform -> broadcast row/col/w loads).
// -------------------------------------------------------------------------
__device__ __forceinline__ void atom_add_f32(float* p, float v)
{
#if defined(__HIP_DEVICE_COMPILE__)
    unsafeAtomicAdd(p, v);     // lowers to global_atomic_add_f32
#else
    atomicAdd(p, v);
#endif
}

__global__ void __launch_bounds__(256)
gcn_spmm_scatter4(const int* __restrict__ erow, const int* __restrict__ ecol,
                  const float* __restrict__ ew, const float* __restrict__ src,
                  float* __restrict__ dst, long long total4, int qshift)
{
    long long t = (long long)blockIdx.x * blockDim.x + threadIdx.x;
    if (t >= total4) return;
    const int Q  = 1 << qshift;              // D/4
    const long long e = t >> qshift;
    const int f4 = (int)(t & (Q - 1));

    const int   r  = erow[e];
    const int   c  = ecol[e];
    const float wv = ew[e];

    const float4 s = ((const float4*)(src + ((size_t)c << (qshift + 2))))[f4];
    float* d = dst + ((size_t)r << (qshift + 2)) + 4 * f4;
    atom_add_f32(d + 0, wv * s.x);
    atom_add_f32(d + 1, wv * s.y);
    atom_add_f32(d + 2, wv * s.z);
    atom_add_f32(d + 3, wv * s.w);
}

// -------------------------------------------------------------------------
// Elementwise helpers (float4 wide)
// -------------------------------------------------------------------------
__global__ void __launch_bounds__(256)
gcn_zero_f4(float4* __restrict__ p, long long n4)
{
    long long t = (long long)blockIdx.x * blockDim.x + threadIdx.x;
    if (t < n4) p[t] = make_float4(0.f, 0.f, 0.f, 0.f);
}

__global__ void __launch_bounds__(256)
gcn_bias_relu4(float4* __restrict__ y, const float4* __restrict__ b,
               long long total4, int qshift)
{
    long long t = (long long)blockIdx.x * blockDim.x + threadIdx.x;
    if (t >= total4) return;
    const int f4 = (int)(t & ((1 << qshift) - 1));
    const float4 bb = b[f4];
    float4 v = y[t];
    v.x = fmaxf(v.x + bb.x, 0.f);
    v.y = fmaxf(v.y + bb.y, 0.f);
    v.z = fmaxf(v.z + bb.z, 0.f);
    v.w = fmaxf(v.w + bb.w, 0.f);
    y[t] = v;
}

// -------------------------------------------------------------------------
// Launch: x@W1 -> spmm -> +b1,relu -> @W2 -> spmm -> +b2,relu
// -------------------------------------------------------------------------
extern "C" void kernel_launch(void* const* d_in, const int* in_sizes, int n_in,
                              void* d_out, int out_size, void* d_ws, size_t ws_size,
                              hipStream_t stream)
{
    const float* x  = (const float*)d_in[0];
    const int*   er = (const int*)  d_in[1];
    const int*   ec = (const int*)  d_in[2];
    const float* ew = (const float*)d_in[3];
    const float* W1 = (const float*)d_in[4];
    const float* b1 = (const float*)d_in[5];
    const float* W2 = (const float*)d_in[6];
    const float* b2 = (const float*)d_in[7];

    const int E  = in_sizes[1];          // 800000
    const int D1 = in_sizes[5];          // 128
    const int D2 = in_sizes[7];          // 32
    const int D0 = in_sizes[4] / D1;     // 256
    const int N  = in_sizes[0] / D0;     // 50000
    const int q1 = __builtin_ctz(D1) - 2;   // 5  (D1/4 groups)
    const int q2 = __builtin_ctz(D2) - 2;   // 3  (D2/4 groups)

    float* s1   = (float*)d_ws;                 // [N, D1] = x @ W1
    float* hacc = s1 + (size_t)N * D1;          // [N, D1] scatter accumulator -> h
    float* s2   = s1;                           // [N, D2] reuses s1 region
    float* oacc = (float*)d_out;                // [N, D2] final accumulator

    const int THR = 256;

    // 1. s1 = x @ W1   (NT = 128/16 = 8 tiles per wave, KC = 64)
    {
        long long waves  = (N + 15) / 16;
        int blocks = (int)((waves + 7) / 8);
        gcn_gemm_wmma_lds<8, 64><<<blocks, THR, 0, stream>>>(x, W1, s1, N, D0);
    }
    // 2. hacc = 0
    {
        long long n4 = (long long)N * D1 / 4;
        gcn_zero_f4<<<(int)((n4 + THR - 1) / THR), THR, 0, stream>>>((float4*)hacc, n4);
    }
    // 3. hacc += A_sparse @ s1
    {
        long long total4 = (long long)E << q1;
        gcn_spmm_scatter4<<<(int)((total4 + THR - 1) / THR), THR, 0, stream>>>(
            er, ec, ew, s1, hacc, total4, q1);
    }
    // 4. h = relu(hacc + b1)   (in place)
    {
        long long total4 = (long long)N * D1 / 4;
        gcn_bias_relu4<<<(int)((total4 + THR - 1) / THR), THR, 0, stream>>>(
            (float4*)hacc, (const float4*)b1, total4, q1);
    }
    // 5. s2 = h @ W2   (NT = 32/16 = 2 tiles per wave, KC = 64)
    {
        long long waves  = (N + 15) / 16;
        int blocks = (int)((waves + 7) / 8);
        gcn_gemm_wmma_lds<2, 64><<<blocks, THR, 0, stream>>>(hacc, W2, s2, N, D1);
    }
    // 6. oacc = 0
    {
        long long n4 = (long long)N * D2 / 4;
        gcn_zero_f4<<<(int)((n4 + THR - 1) / THR), THR, 0, stream>>>((float4*)oacc, n4);
    }
    // 7. oacc += A_sparse @ s2
    {
        long long total4 = (long long)E << q2;
        gcn_spmm_scatter4<<<(int)((total4 + THR - 1) / THR), THR, 0, stream>>>(
            er, ec, ew, s2, oacc, total4, q2);
    }
    // 8. out = relu(oacc + b2)  (in place)
    {
        long long total4 = (long long)N * D2 / 4;
        gcn_bias_relu4<<<(int)((total4 + THR - 1) / THR), THR, 0, stream>>>(
            (float4*)oacc, (const float4*)b2, total4, q2);
    }
}